// BaseRouter_3435973837295
// MI455X (gfx1250) — compile-verified
//
#include <hip/hip_runtime.h>
#include <hip/hip_bf16.h>

// ---------------------------------------------------------------------------
// MoE router for MI455X (gfx1250, wave32).
// B=2, S=2048 -> N=4096 tokens, H=1024, E=8, TOPK=2, CAPACITY=1536.
// Output layout (flat fp32): dispatch[N,E,CAP] | combine[N,E,CAP] | probs[N,E] | aux[1]
//
// Roofline: 403 MB of (mostly zero) routing-tensor stores dominate
// (~17 us @ 23.3 TB/s) -> written with non-temporal b128 stores so the
// 192 MB L2 is not swept twice by zeros. GEMM1 (8.6 GFLOP) uses full-precision
// V_WMMA_F32_16X16X4_F32 (bf16 WMMA would be ~8x faster per-instruction but
// risks flipping near-tie top-2 expert picks, corrupting whole dispatch rows,
// and GEMM1 already hides under the store phase). Each wave computes a 16x64
// tile: one A fetch feeds 4 independent WMMA chains (6.5 FLOP/B from cache,
// 4-deep WMMA ILP to hide load latency).
// ---------------------------------------------------------------------------

#define N_TOK 4096
#define HDIM  1024
#define NEXP  8
#define CAP   1536
#define NEC   (N_TOK * NEXP * CAP)   // 50,331,648 elements per routing tensor

typedef float v2f __attribute__((ext_vector_type(2)));
typedef float v4f __attribute__((ext_vector_type(4)));
typedef float v8f __attribute__((ext_vector_type(8)));

// ---------------------------------------------------------------------------
// Kernel 1: zero-fill dispatch+combine (403 MB) with NON-TEMPORAL b128 stores
// (TH=NT: streams past L2, which these zeros would otherwise flush twice).
// ---------------------------------------------------------------------------
__global__ __launch_bounds__(256) void zero_kernel(v4f* __restrict__ p, long n4) {
    long i      = (long)blockIdx.x * blockDim.x + threadIdx.x;
    long stride = (long)gridDim.x * blockDim.x;
    v4f z = {0.f, 0.f, 0.f, 0.f};
    for (; i < n4; i += stride)
        __builtin_nontemporal_store(z, p + i);
}

// ---------------------------------------------------------------------------
// Kernel 2: h = relu(X @ W1 + b1) with V_WMMA_F32_16X16X4_F32.
// Block = 256 threads = 8 waves; block tile 16 x 512, wave tile 16 x 64
// (4 accumulators reuse each A fetch; 256 K-steps, 1024 WMMA per wave).
//
// Per-lane WMMA f32 layouts (ISA 7.12.2):
//   A 16x4:  lane l -> row M = l&15, Ks (l>>4)*2 .. +1  (2 VGPRs)
//   B 4x16:  lane l -> col N = l&15, Ks (l>>4)*2 .. +1  (symmetric)
//   C/D:     VGPR r, lanes 0-15: M=r,N=lane ; lanes 16-31: M=r+8,N=lane-16
// ---------------------------------------------------------------------------
__global__ __launch_bounds__(256) void gemm1_relu_kernel(
        const float* __restrict__ X, const float* __restrict__ W1,
        const float* __restrict__ b1, float* __restrict__ Hout) {
    const int lane = threadIdx.x & 31;
    const int wave = threadIdx.x >> 5;
    const int m0 = blockIdx.x * 16;
    const int n0 = blockIdx.y * 512 + wave * 64;
    const int mr = lane & 15;
    const int kb = (lane >> 4) << 1;   // 0 or 2

    const float* arow  = X  + (size_t)(m0 + mr) * HDIM + kb;   // A: row m, K pair
    const float* bbase = W1 + (size_t)kb * HDIM + n0 + mr;     // B: col n, K pair

    v8f acc0 = {0.f,0.f,0.f,0.f,0.f,0.f,0.f,0.f};
    v8f acc1 = acc0, acc2 = acc0, acc3 = acc0;

#pragma unroll 4
    for (int k = 0; k < HDIM; k += 4) {
        v2f a; a.x = arow[k]; a.y = arow[k + 1];
        const float* r0 = bbase + (size_t)k * HDIM;          // row k+kb
        const float* r1 = r0 + HDIM;                         // row k+kb+1
        v2f b0; b0.x = r0[0];  b0.y = r1[0];
        v2f b1v; b1v.x = r0[16]; b1v.y = r1[16];
        v2f b2v; b2v.x = r0[32]; b2v.y = r1[32];
        v2f b3v; b3v.x = r0[48]; b3v.y = r1[48];
        acc0 = __builtin_amdgcn_wmma_f32_16x16x4_f32(false, a, false, b0,  (short)0, acc0, false, false);
        acc1 = __builtin_amdgcn_wmma_f32_16x16x4_f32(false, a, false, b1v, (short)0, acc1, false, false);
        acc2 = __builtin_amdgcn_wmma_f32_16x16x4_f32(false, a, false, b2v, (short)0, acc2, false, false);
        acc3 = __builtin_amdgcn_wmma_f32_16x16x4_f32(false, a, false, b3v, (short)0, acc3, false, false);
    }

    const int rbase = m0 + ((lane >> 4) << 3);
    v8f accs[4] = {acc0, acc1, acc2, acc3};
#pragma unroll
    for (int j = 0; j < 4; ++j) {
        const int col   = n0 + j * 16 + mr;
        const float bias = b1[col];
#pragma unroll
        for (int r = 0; r < 8; ++r) {
            float v = accs[j][r] + bias;
            Hout[(size_t)(rbase + r) * HDIM + col] = v > 0.f ? v : 0.f;
        }
    }
}

// ---------------------------------------------------------------------------
// Kernel 3: logits = h @ W2 + b2; softmax over E=8; top-2; renormalize.
// One wave32 per token (8 waves / block). Memory-bound on h (16.8 MB).
// ---------------------------------------------------------------------------
__global__ __launch_bounds__(256) void router_kernel(
        const float* __restrict__ Hm, const float* __restrict__ W2,
        const float* __restrict__ b2, float* __restrict__ probs,
        int* __restrict__ tki, float* __restrict__ tkp) {
    const int lane = threadIdx.x & 31;
    const int wave = threadIdx.x >> 5;
    const int t = blockIdx.x * 8 + wave;

    const float* hrow = Hm + (size_t)t * HDIM;
    float acc[NEXP];
#pragma unroll
    for (int e = 0; e < NEXP; ++e) acc[e] = 0.f;

    for (int k = lane; k < HDIM; k += 32) {
        float hv = hrow[k];
        const float* wrow = W2 + (size_t)k * NEXP;
#pragma unroll
        for (int e = 0; e < NEXP; ++e) acc[e] = fmaf(hv, wrow[e], acc[e]);
    }

    // wave32 butterfly reduction: every lane ends with full logits
#pragma unroll
    for (int e = 0; e < NEXP; ++e) {
        float v = acc[e];
#pragma unroll
        for (int off = 16; off > 0; off >>= 1) v += __shfl_xor(v, off, 32);
        acc[e] = v + b2[e];
    }

    // softmax
    float mx = acc[0];
#pragma unroll
    for (int e = 1; e < NEXP; ++e) mx = fmaxf(mx, acc[e]);
    float sum = 0.f;
#pragma unroll
    for (int e = 0; e < NEXP; ++e) { acc[e] = expf(acc[e] - mx); sum += acc[e]; }
    float inv = 1.f / sum;
#pragma unroll
    for (int e = 0; e < NEXP; ++e) acc[e] *= inv;

    if (lane < NEXP) probs[(size_t)t * NEXP + lane] = acc[lane];

    if (lane == 0) {
        // top-2, lowest-index-first tie break (matches jax.lax.top_k)
        int i0 = 0; float p0 = acc[0];
#pragma unroll
        for (int e = 1; e < NEXP; ++e) if (acc[e] > p0) { p0 = acc[e]; i0 = e; }
        int i1 = -1; float p1 = -1.f;
#pragma unroll
        for (int e = 0; e < NEXP; ++e) if (e != i0 && acc[e] > p1) { p1 = acc[e]; i1 = e; }
        float s = p0 + p1;
        tki[t * 2]     = i0;
        tki[t * 2 + 1] = i1;
        tkp[t * 2]     = p0 / s;
        tkp[t * 2 + 1] = p1 / s;
    }
}

// ---------------------------------------------------------------------------
// Kernel 4: deterministic dispatch/combine scatter. One block per expert.
// Reproduces the reference's sequential counter: slot rank ordered by
// (topk-slot i major, token order). LDS Hillis-Steele prefix scan per chunk.
// ---------------------------------------------------------------------------
__global__ __launch_bounds__(256) void scatter_kernel(
        const int* __restrict__ tki, const float* __restrict__ tkp,
        float* __restrict__ dispatch, float* __restrict__ combine) {
    __shared__ int s[256];
    const int e   = blockIdx.x;
    const int tid = threadIdx.x;
    int running = 0;

    for (int i = 0; i < 2; ++i) {
        for (int base = 0; base < N_TOK; base += 256) {
            const int t = base + tid;
            const int match = (tki[t * 2 + i] == e) ? 1 : 0;
            s[tid] = match;
            __syncthreads();
            // inclusive scan over 256 flags
            for (int off = 1; off < 256; off <<= 1) {
                int v = s[tid];
                int u = (tid >= off) ? s[tid - off] : 0;
                __syncthreads();
                s[tid] = v + u;
                __syncthreads();
            }
            const int incl  = s[tid];
            const int total = s[255];
            __syncthreads();
            if (match) {
                int slot = running + incl - 1;          // exclusive rank
                if (slot < CAP) {
                    size_t o = (size_t)t * (NEXP * CAP) + (size_t)e * CAP + slot;
                    dispatch[o] = 1.0f;
                    combine[o]  = tkp[t * 2 + i];
                }
            }
            running += total;
        }
    }
}

// ---------------------------------------------------------------------------
// Kernel 5: aux loss. Single block, deterministic LDS tree reduction.
// ---------------------------------------------------------------------------
__global__ __launch_bounds__(1024) void aux_kernel(
        const float* __restrict__ probs, float* __restrict__ aux) {
    __shared__ float red[1024 * NEXP];   // 32 KB of 320 KB LDS
    const int tid = threadIdx.x;
    float pe[NEXP];
#pragma unroll
    for (int e = 0; e < NEXP; ++e) pe[e] = 0.f;
    for (int t = tid; t < N_TOK; t += 1024) {
#pragma unroll
        for (int e = 0; e < NEXP; ++e) pe[e] += probs[(size_t)t * NEXP + e];
    }
#pragma unroll
    for (int e = 0; e < NEXP; ++e) red[tid * NEXP + e] = pe[e];
    __syncthreads();
    for (int st = 512; st > 0; st >>= 1) {
        if (tid < st) {
#pragma unroll
            for (int e = 0; e < NEXP; ++e)
                red[tid * NEXP + e] += red[(tid + st) * NEXP + e];
        }
        __syncthreads();
    }
    if (tid == 0) {
        float a = 0.f;
#pragma unroll
        for (int e = 0; e < NEXP; ++e) {
            float p = red[e] * (1.0f / (float)N_TOK);
            a += p * logf(p * (float)NEXP + 1e-9f);
        }
        aux[0] = a;
    }
}

// ---------------------------------------------------------------------------
// Launch. Inputs: hidden_states, W1, b1, W2, b2 (all fp32).
// Workspace: h (16.8 MB) | topk_i (32 KB) | topk_p (32 KB).
// ---------------------------------------------------------------------------
extern "C" void kernel_launch(void* const* d_in, const int* in_sizes, int n_in,
                              void* d_out, int out_size, void* d_ws, size_t ws_size,
                              hipStream_t stream) {
    const float* X  = (const float*)d_in[0];
    const float* W1 = (const float*)d_in[1];
    const float* b1 = (const float*)d_in[2];
    const float* W2 = (const float*)d_in[3];
    const float* b2 = (const float*)d_in[4];

    float* out      = (float*)d_out;
    float* dispatch = out;
    float* combine  = out + (size_t)NEC;
    float* probs    = out + 2 * (size_t)NEC;
    float* aux      = probs + (size_t)N_TOK * NEXP;

    char* ws   = (char*)d_ws;
    float* Hm  = (float*)ws;                                        // 4096*1024*4 B
    int*   tki = (int*)(ws + (size_t)N_TOK * HDIM * sizeof(float));
    float* tkp = (float*)(ws + (size_t)N_TOK * HDIM * sizeof(float)
                             + (size_t)N_TOK * 2 * sizeof(int));

    // 1) zero dispatch+combine (403 MB) — NT streaming stores
    long n4 = (2L * (long)NEC) / 4;
    zero_kernel<<<4096, 256, 0, stream>>>((v4f*)out, n4);

    // 2) h = relu(X @ W1 + b1) via f32 WMMA, 16x64 per wave
    gemm1_relu_kernel<<<dim3(N_TOK / 16, HDIM / 512), 256, 0, stream>>>(X, W1, b1, Hm);

    // 3) logits/softmax/top-2 (one wave per token)
    router_kernel<<<N_TOK / 8, 256, 0, stream>>>(Hm, W2, b2, probs, tki, tkp);

    // 4) deterministic capacity-ordered scatter
    scatter_kernel<<<NEXP, 256, 0, stream>>>(tki, tkp, dispatch, combine);

    // 5) aux loss
    aux_kernel<<<1, 1024, 0, stream>>>(probs, aux);
}